// EctLayer_39994735461117
// MI455X (gfx1250) — compile-verified
//
#include <hip/hip_runtime.h>

// ---------------------------------------------------------------------------
// EctLayer for MI455X (gfx1250, wave32).
//
// Roofline: naive cost = N*S*T = 204.8M sigmoids + scatter adds; bytes are
// trivial (2.6 MB in, 2 MB out) -> transcendental/atomic bound, not HBM.
// SHARP=500 vs lin step 2/63 makes the sigmoid a near-step: only
// |500*(lin-nh)| < 18 (~2-3 s values per (n,t)) need real math; the
// saturated tail is a "+1 for all s >= s_hi" step counter resolved by a
// prefix sum at flush. nh = x@v uses V_WMMA_F32_16X16X4_F32
// (16 nodes x 16 thetas per instruction, K=3 zero-padded to 4).
// ---------------------------------------------------------------------------

typedef __attribute__((ext_vector_type(2))) float v2f;
typedef __attribute__((ext_vector_type(8))) float v8f;

#define TDIM   64          // thetas
#define SDIM   64          // bump steps
#define HPAD   65          // LDS row stride (pad to avoid 32-way bank conflicts)
#define SHARPF 500.0f
#define CUTZ   18.0f       // |z| >= 18 -> sigmoid rounds to 0/1 in fp32
#define LOG2E  1.44269504f

// ---------------------------------------------------------------------------
// Kernel 0: zero the output (harness poisons d_out with 0xAA).
// ---------------------------------------------------------------------------
__global__ void ect_zero_kernel(float* __restrict__ out, int n) {
    int i = blockIdx.x * blockDim.x + threadIdx.x;
    if (i < n) out[i] = 0.0f;
}

// ---------------------------------------------------------------------------
// Kernel 1: nh[n][t] = sum_f x[n][f] * v[f][t] via V_WMMA_F32_16X16X4_F32.
// One wave handles one 16(node) x 16(theta) tile. K=3 padded to 4 with zeros.
// ISA 7.12.2 f32 layouts:
//   A (16x4): lane m=l%16 is row M; lanes 0-15 hold K={0,1} in VGPR{0,1},
//             lanes 16-31 hold K={2,3}.
//   B (4x16): lane n=l%16 is col N; VGPR0 holds rows K=0 (lanes 0-15) /
//             K=2 (lanes 16-31); VGPR1 holds K=1 / K=3.
//   C/D (16x16): lane l%16 is col N; VGPR r is row M=r (lanes 0-15) or
//             M=r+8 (lanes 16-31).
// Operand setup is branchless (loads + v_cndmask), and the full-tile store
// path is wave-uniform straight-line: one address register, 8 stores at
// immediate offsets r*T*4 bytes (D rows are contiguous in n -> stride 256B).
// ---------------------------------------------------------------------------
__global__ void nh_wmma_kernel(const float* __restrict__ x,
                               const float* __restrict__ v,
                               float* __restrict__ nh,
                               int Nn, int nTiles, int tTiles) {
    int wave  = blockIdx.x * (blockDim.x >> 5) + (threadIdx.x >> 5);
    if (wave >= nTiles * tTiles) return;          // whole wave exits together
    int ntile = wave / tTiles;
    int ttile = wave - ntile * tTiles;

    int l  = threadIdx.x & 31;
    int m  = l & 15;
    int hi = l >> 4;                              // 0: lanes 0-15, 1: lanes 16-31

    int node  = ntile * 16 + m;
    int nodeC = min(node, Nn - 1);                // clamp loads (remainder tile)

    // A operands: lanes<16 want (f0,f1); lanes>=16 want (f2,0). Branchless:
    float a0 = x[nodeC * 3 + (hi << 1)];          // f0 or f2
    float a1 = x[nodeC * 3 + 1];                  // f1 (selected below)
    v2f a;
    a.x = a0;
    a.y = hi ? 0.0f : a1;

    int tcol = ttile * 16 + m;
    float b0 = v[(hi << 1) * TDIM + tcol];        // row K=0 or K=2
    float b1 = v[1 * TDIM + tcol];                // row K=1 (selected below)
    v2f b;
    b.x = b0;
    b.y = hi ? 0.0f : b1;

    v8f c = {};
    c = __builtin_amdgcn_wmma_f32_16x16x4_f32(
            /*neg_a=*/false, a, /*neg_b=*/false, b,
            /*c_mod=*/(short)0, c, /*reuse_a=*/false, /*reuse_b=*/false);

    int rowBase = ntile * 16 + hi * 8;
    float* p = nh + (size_t)rowBase * TDIM + tcol;
    if (ntile * 16 + 16 <= Nn) {
        // full tile (always for N % 16 == 0): straight-line stores,
        // shared address register + immediate offsets (stride T*4 = 256B)
#pragma unroll
        for (int r = 0; r < 8; ++r) p[r * TDIM] = c[r];
    } else {
#pragma unroll
        for (int r = 0; r < 8; ++r)
            if (rowBase + r < Nn) p[r * TDIM] = c[r];
    }
}

// ---------------------------------------------------------------------------
// Kernel 2: banded sigmoid accumulation + segment flush.
// Each block owns a contiguous node chunk; batch is sorted, so segments are
// contiguous. Per segment: accumulate into LDS tiles
//   Hb[s][t] : explicit sigmoid values in the transition band
//   Cb[s][t] : "+1 for all s' >= s" step counters
// then flush with a prefix-sum over s and global f32 atomics.
// Thread mapping: t = tid & 63 (coalesced nh reads, padded LDS banks),
// blockDim/64 nodes in flight (tid >> 6).
// ---------------------------------------------------------------------------
__global__ void ect_accum_kernel(const float* __restrict__ nh,
                                 const float* __restrict__ lin,
                                 const int*  __restrict__ batch,
                                 float* __restrict__ out,
                                 int Nn, int chunk) {
    __shared__ float Hb[SDIM * HPAD];
    __shared__ float Cb[SDIM * HPAD];
    __shared__ float linS[SDIM];

    int tid = threadIdx.x;
    int n0  = blockIdx.x * chunk;
    int n1  = min(n0 + chunk, Nn);
    if (n0 >= n1) return;

    if (tid < SDIM) linS[tid] = lin[tid];
    __syncthreads();

    float lin0     = linS[0];
    float invDelta = 1.0f / (linS[1] - lin0);             // = 31.5 for S=64, R=1
    float cutSteps = (CUTZ / SHARPF) * invDelta;          // ~1.134 steps

    int t       = tid & (TDIM - 1);
    int nodeOff = tid >> 6;
    int nodeStp = blockDim.x >> 6;                        // 4 for 256 threads

    int seg = n0;
    while (seg < n1) {
        int b = batch[seg];
        // upper_bound(batch, b) in [seg+1, n1) -- redundant per-thread, cheap
        int lo2 = seg + 1, hi2 = n1;
        while (lo2 < hi2) {
            int mid = (lo2 + hi2) >> 1;
            if (batch[mid] == b) lo2 = mid + 1; else hi2 = mid;
        }
        int segEnd = lo2;

        for (int i = tid; i < SDIM * HPAD; i += blockDim.x) {
            Hb[i] = 0.0f; Cb[i] = 0.0f;
        }
        __syncthreads();

        for (int n = seg + nodeOff; n < segEnd; n += nodeStp) {
            float nhv  = nh[(size_t)n * TDIM + t];
            float base = (nhv - lin0) * invDelta;
            int sLo = (int)ceilf(base - cutSteps);        // first s with z > -CUT
            int sHi = (int)ceilf(base + cutSteps);        // first s with z >= +CUT
            sLo = min(max(sLo, 0), SDIM);
            sHi = min(max(sHi, 0), SDIM);
            if (sHi < SDIM) atomicAdd(&Cb[sHi * HPAD + t], 1.0f);  // ds_add_f32
            for (int s = sLo; s < sHi; ++s) {
                float z   = SHARPF * (linS[s] - nhv);
                float e   = __builtin_amdgcn_exp2f(-z * LOG2E);    // v_exp_f32
                float sig = __builtin_amdgcn_rcpf(1.0f + e);       // v_rcp_f32
                atomicAdd(&Hb[s * HPAD + t], sig);
            }
        }
        __syncthreads();

        if (tid < TDIM) {
            float run   = 0.0f;
            float* outB = out + (size_t)b * SDIM * TDIM;
            for (int s = 0; s < SDIM; ++s) {
                run += Cb[s * HPAD + tid];                 // prefix over steps
                float val = Hb[s * HPAD + tid] + run;
                if (val != 0.0f)
                    unsafeAtomicAdd(&outB[s * TDIM + tid], val);   // global_atomic_add_f32
            }
        }
        __syncthreads();
        seg = segEnd;
    }
}

// ---------------------------------------------------------------------------
// Launch: inputs in setup_inputs() order: x (N*3 f32), v (3*64 f32),
// lin (64 f32), batch (N i32). Output: B*64*64 f32. d_ws holds nh (N*64 f32).
// ---------------------------------------------------------------------------
extern "C" void kernel_launch(void* const* d_in, const int* in_sizes, int n_in,
                              void* d_out, int out_size, void* d_ws, size_t ws_size,
                              hipStream_t stream) {
    const float* x     = (const float*)d_in[0];
    const float* v     = (const float*)d_in[1];
    const float* lin   = (const float*)d_in[2];
    const int*   batch = (const int*)d_in[3];
    float*       out   = (float*)d_out;
    float*       nh    = (float*)d_ws;   // N*T*4 = 12.8 MB scratch

    int Nn = in_sizes[0] / 3;            // 50000
    int T  = in_sizes[1] / 3;            // 64

    // 0) zero output (accumulated with atomics below)
    ect_zero_kernel<<<(out_size + 255) / 256, 256, 0, stream>>>(out, out_size);

    // 1) nh = x @ v via WMMA f32 16x16x4; 8 waves (tiles) per 256-thread block
    int nTiles = (Nn + 15) / 16;
    int tTiles = T / 16;
    int totalTiles = nTiles * tTiles;
    nh_wmma_kernel<<<(totalTiles + 7) / 8, 256, 0, stream>>>(x, v, nh, Nn, nTiles, tTiles);

    // 2) banded accumulation + segment flush
    const int CHUNK = 256;
    ect_accum_kernel<<<(Nn + CHUNK - 1) / CHUNK, CHUNK, 0, stream>>>(
        nh, lin, batch, out, Nn, CHUNK);
}